// SimpleGNN_37117107372163
// MI455X (gfx1250) — compile-verified
//
#include <hip/hip_runtime.h>
#include <hip/hip_bf16.h>

typedef __attribute__((ext_vector_type(2))) float v2f;
typedef __attribute__((ext_vector_type(8))) float v8f;

#define HID 32
#define KMAX 128   // max inner dim (IN_DIM = 128, hidden = 32)

// ---------------------------------------------------------------------------
// Degree / normalization (integer atomics, then rsqrt in place)
// ---------------------------------------------------------------------------
__global__ void deg_init_kernel(unsigned int* __restrict__ deg, int N) {
    int i = blockIdx.x * blockDim.x + threadIdx.x;
    if (i < N) deg[i] = 1u;   // self-loop contribution
}

__global__ void deg_accum_kernel(const int* __restrict__ col,
                                 unsigned int* __restrict__ deg, int E) {
    int e = blockIdx.x * blockDim.x + threadIdx.x;
    if (e < E) atomicAdd(&deg[col[e]], 1u);
}

__global__ void deg_rsqrt_kernel(float* __restrict__ dinv, int N) {
    int i = blockIdx.x * blockDim.x + threadIdx.x;
    if (i < N) {
        unsigned int d = ((const unsigned int*)dinv)[i];
        dinv[i] = rsqrtf((float)d);   // d >= 1 always
    }
}

// ---------------------------------------------------------------------------
// GEMM + self-loop init:  hW = H @ W ;  agg = hW * dinv^2
// V_WMMA_F32_16X16X4_F32. One wave: 16 rows x 32 cols. 8 waves/block = 128 rows.
// W is staged transposed+padded in LDS so each B-frag is one ds_load_b64.
// ---------------------------------------------------------------------------
__global__ void gcn_gemm_wmma_kernel(const float* __restrict__ H,
                                     const float* __restrict__ W,
                                     const float* __restrict__ dinv,
                                     float* __restrict__ Y,
                                     float* __restrict__ A,
                                     int N, int K) {
    __shared__ float Wt[HID * (KMAX + 4)];   // [n][k], pad 4 for bank spread

    // cooperative transpose of W[K][32] -> Wt[n][k]
    for (int idx = threadIdx.x; idx < K * HID; idx += blockDim.x) {
        int kk = idx >> 5;      // W row-major: idx = kk*32 + n
        int n  = idx & 31;
        Wt[n * (K + 4) + kk] = W[idx];
    }
    __syncthreads();

    const int lane  = threadIdx.x & 31;
    const int wave  = threadIdx.x >> 5;                 // 0..7
    const int tileM = blockIdx.x * 128 + wave * 16;
    const int m     = lane & 15;                        // M (A) / N (B) index
    const int half  = lane >> 4;                        // K-pair selector

    const int   row   = tileM + m;
    const float scale = (row < N) ? 1.0f : 0.0f;
    const int   rowC  = (row < N) ? row : (N - 1);
    const float* __restrict__ Hrow = H + (size_t)rowC * (size_t)K + half * 2;
    const float* __restrict__ WtA  = Wt + (size_t)m        * (K + 4) + half * 2;
    const float* __restrict__ WtB  = Wt + (size_t)(m + 16) * (K + 4) + half * 2;

    v8f c0 = {};   // output columns 0..15
    v8f c1 = {};   // output columns 16..31

#pragma unroll 4
    for (int k = 0; k < K; k += 4) {
        v2f a = *(const v2f*)(Hrow + k);    // global_load_b64
        a.x *= scale;
        a.y *= scale;
        v2f b0 = *(const v2f*)(WtA + k);    // ds_load_b64
        v2f b1 = *(const v2f*)(WtB + k);    // ds_load_b64

        c0 = __builtin_amdgcn_wmma_f32_16x16x4_f32(false, a, false, b0,
                                                   (short)0, c0, false, false);
        c1 = __builtin_amdgcn_wmma_f32_16x16x4_f32(false, a, false, b1,
                                                   (short)0, c1, false, false);
    }

    // C/D layout: VGPR v -> M = v + half*8, N = m (c0) / m+16 (c1)
#pragma unroll
    for (int v = 0; v < 8; ++v) {
        const int r = tileM + v + half * 8;
        if (r < N) {
            const float di  = dinv[r];
            const float di2 = di * di;
            const float s0  = c0[v];
            const float s1  = c1[v];
            Y[(size_t)r * HID + m]      = s0;
            Y[(size_t)r * HID + m + 16] = s1;
            A[(size_t)r * HID + m]      = s0 * di2;   // self-loop term
            A[(size_t)r * HID + m + 16] = s1 * di2;
        }
    }
}

// ---------------------------------------------------------------------------
// Edge scatter: one wave per edge, lane = channel (HID == wave32 width).
// agg[col][lane] += hW[row][lane] * dinv[row]*dinv[col]
// ---------------------------------------------------------------------------
__global__ void edge_scatter_kernel(const float* __restrict__ hW,
                                    const float* __restrict__ dinv,
                                    const int* __restrict__ row,
                                    const int* __restrict__ col,
                                    float* __restrict__ agg, int E) {
    const int lane = threadIdx.x & 31;
    const int e    = (blockIdx.x * blockDim.x + threadIdx.x) >> 5;
    if (e >= E) return;
    const int r = row[e];
    const int c = col[e];
    const float nrm = dinv[r] * dinv[c];
    const float v = hW[(size_t)r * HID + lane] * nrm;
    atomicAdd(&agg[(size_t)c * HID + lane], v);
}

// ---------------------------------------------------------------------------
// h[i][j] = relu(agg[i][j] + b[j])
// ---------------------------------------------------------------------------
__global__ void bias_relu_kernel(const float* __restrict__ agg,
                                 const float* __restrict__ b,
                                 float* __restrict__ h, int N) {
    int i = blockIdx.x * blockDim.x + threadIdx.x;
    if (i >= N * HID) return;
    float v = agg[i] + b[i & 31];
    h[i] = v > 0.0f ? v : 0.0f;
}

// ---------------------------------------------------------------------------
// Pooling
// ---------------------------------------------------------------------------
__global__ void pool_zero_kernel(float* __restrict__ pooled,
                                 float* __restrict__ cnt, int G) {
    int i = blockIdx.x * blockDim.x + threadIdx.x;
    if (i < G * HID) pooled[i] = 0.0f;
    if (i < G)       cnt[i]    = 0.0f;
}

__global__ void pool_accum_kernel(const float* __restrict__ h,
                                  const int* __restrict__ batch,
                                  float* __restrict__ pooled,
                                  float* __restrict__ cnt, int N) {
    const int lane = threadIdx.x & 31;
    const int node = (blockIdx.x * blockDim.x + threadIdx.x) >> 5;
    if (node >= N) return;
    const int g = batch[node];
    atomicAdd(&pooled[(size_t)g * HID + lane], h[(size_t)node * HID + lane]);
    if (lane == 0) atomicAdd(&cnt[g], 1.0f);
}

__global__ void pool_final_kernel(const float* __restrict__ pooled,
                                  const float* __restrict__ cnt,
                                  const float* __restrict__ lin_w,
                                  const float* __restrict__ lin_b,
                                  float* __restrict__ out, int G) {
    const int lane = threadIdx.x & 31;
    const int g    = (blockIdx.x * blockDim.x + threadIdx.x) >> 5;
    if (g >= G) return;
    float c = cnt[g];
    c = c > 1.0f ? c : 1.0f;
    float v = (pooled[(size_t)g * HID + lane] / c) * lin_w[lane];
#pragma unroll
    for (int off = 16; off > 0; off >>= 1)
        v += __shfl_xor(v, off, 32);
    if (lane == 0) out[g] = v + lin_b[0];
}

// ---------------------------------------------------------------------------
// Host launch
// ---------------------------------------------------------------------------
extern "C" void kernel_launch(void* const* d_in, const int* in_sizes, int n_in,
                              void* d_out, int out_size, void* d_ws, size_t ws_size,
                              hipStream_t stream) {
    const float* x     = (const float*)d_in[0];   // [N, IN_DIM]
    const float* W1    = (const float*)d_in[1];   // [IN_DIM, 32]
    const float* Wh    = (const float*)d_in[2];   // [3, 32, 32]
    const float* bs    = (const float*)d_in[3];   // [4, 32]
    const float* lin_w = (const float*)d_in[4];   // [32]
    const float* lin_b = (const float*)d_in[5];   // [1]
    const int*   eidx  = (const int*)d_in[6];     // [2, E]
    const int*   batch = (const int*)d_in[7];     // [N]

    const int N  = in_sizes[7];
    const int E  = in_sizes[6] / 2;
    const int IN = in_sizes[0] / N;
    const int G  = out_size;

    const int* row = eidx;          // sources
    const int* col = eidx + E;      // targets

    // Workspace layout (floats)
    float* ws     = (float*)d_ws;
    float* dinv   = ws;                           // N (also int deg scratch)
    float* hW     = dinv + N;                     // N*32
    float* agg    = hW  + (size_t)N * HID;        // N*32
    float* hbuf   = agg + (size_t)N * HID;        // N*32
    float* pooled = hbuf + (size_t)N * HID;       // G*32
    float* cnt    = pooled + (size_t)G * HID;     // G

    const int TPB = 256;
    const int nodeElemBlocks = (N * HID + TPB - 1) / TPB;
    const int gemmBlocks     = (N + 127) / 128;
    const int edgeWarpBlocks = (E * 32 + TPB - 1) / TPB;   // one wave per edge

    // --- normalization ---
    deg_init_kernel<<<(N + TPB - 1) / TPB, TPB, 0, stream>>>((unsigned int*)dinv, N);
    deg_accum_kernel<<<(E + TPB - 1) / TPB, TPB, 0, stream>>>(col, (unsigned int*)dinv, E);
    deg_rsqrt_kernel<<<(N + TPB - 1) / TPB, TPB, 0, stream>>>(dinv, N);

    // --- layer 1 (IN -> 32) ---
    gcn_gemm_wmma_kernel<<<gemmBlocks, TPB, 0, stream>>>(x, W1, dinv, hW, agg, N, IN);
    edge_scatter_kernel<<<edgeWarpBlocks, TPB, 0, stream>>>(hW, dinv, row, col, agg, E);
    bias_relu_kernel<<<nodeElemBlocks, TPB, 0, stream>>>(agg, bs + 0 * HID, hbuf, N);

    // --- layers 2..4 (32 -> 32) ---
    for (int l = 0; l < 3; ++l) {
        gcn_gemm_wmma_kernel<<<gemmBlocks, TPB, 0, stream>>>(
            hbuf, Wh + (size_t)l * HID * HID, dinv, hW, agg, N, HID);
        edge_scatter_kernel<<<edgeWarpBlocks, TPB, 0, stream>>>(hW, dinv, row, col, agg, E);
        bias_relu_kernel<<<nodeElemBlocks, TPB, 0, stream>>>(agg, bs + (size_t)(l + 1) * HID, hbuf, N);
    }

    // --- global mean pool + linear ---
    pool_zero_kernel<<<(G * HID + TPB - 1) / TPB, TPB, 0, stream>>>(pooled, cnt, G);
    pool_accum_kernel<<<(N * 32 + TPB - 1) / TPB, TPB, 0, stream>>>(hbuf, batch, pooled, cnt, N);
    pool_final_kernel<<<(G * 32 + TPB - 1) / TPB, TPB, 0, stream>>>(pooled, cnt, lin_w, lin_b,
                                                                    (float*)d_out, G);
}